// MoE_14396730376781
// MI455X (gfx1250) — compile-verified
//
#include <hip/hip_runtime.h>

#define IN_DIM  2048
#define OUT_DIM 2048
#define NEXP    8
#define NBATCH  8192

#define TM 64
#define TN 128
#define KB 16
#define NCHUNK (IN_DIM / KB)

typedef __attribute__((ext_vector_type(2))) float v2f;
typedef __attribute__((ext_vector_type(8))) float v8f;

// ---------------------------------------------------------------------------
// Kernel 0: zero the per-expert token counters
// ---------------------------------------------------------------------------
__global__ void init_counts_kernel(int* __restrict__ counts) {
    if (threadIdx.x < NEXP) counts[threadIdx.x] = 0;
}

// ---------------------------------------------------------------------------
// Kernel 1: gating. One wave32 per token: logits -> softmax -> top-2,
// write dense combine row [B,E] and append token to its two expert lists.
// ---------------------------------------------------------------------------
__global__ __launch_bounds__(256) void gating_kernel(
    const float* __restrict__ x, const float* __restrict__ Wg,
    const float* __restrict__ bg, float* __restrict__ combine,
    int* __restrict__ counts, int* __restrict__ token_idx) {
    const int wave  = threadIdx.x >> 5;
    const int lane  = threadIdx.x & 31;
    const int token = blockIdx.x * 8 + wave;

    float acc[NEXP];
#pragma unroll
    for (int e = 0; e < NEXP; ++e) acc[e] = 0.f;

    const float* xrow = x + (size_t)token * IN_DIM;
    for (int i = lane; i < IN_DIM; i += 32) {
        const float xv = xrow[i];
        const float4* w = (const float4*)(Wg + (size_t)i * NEXP);
        const float4 w0 = w[0], w1 = w[1];
        acc[0] += xv * w0.x; acc[1] += xv * w0.y;
        acc[2] += xv * w0.z; acc[3] += xv * w0.w;
        acc[4] += xv * w1.x; acc[5] += xv * w1.y;
        acc[6] += xv * w1.z; acc[7] += xv * w1.w;
    }
#pragma unroll
    for (int off = 16; off > 0; off >>= 1)
#pragma unroll
        for (int e = 0; e < NEXP; ++e)
            acc[e] += __shfl_down(acc[e], off, 32);

    if (lane == 0) {
        float p[NEXP];
        float mx = -3.0e38f;
#pragma unroll
        for (int e = 0; e < NEXP; ++e) {
            p[e] = acc[e] + bg[e];
            mx = fmaxf(mx, p[e]);
        }
        float sum = 0.f;
#pragma unroll
        for (int e = 0; e < NEXP; ++e) { p[e] = __expf(p[e] - mx); sum += p[e]; }
        const float inv = 1.f / sum;
#pragma unroll
        for (int e = 0; e < NEXP; ++e) p[e] *= inv;

        // top-2 (ties -> lowest index, matching lax.top_k)
        int i1 = 0;
#pragma unroll
        for (int e = 1; e < NEXP; ++e) if (p[e] > p[i1]) i1 = e;
        int i2 = (i1 == 0) ? 1 : 0;
#pragma unroll
        for (int e = 0; e < NEXP; ++e)
            if (e != i1 && e != i2 && p[e] > p[i2]) i2 = e;

        float* crow = combine + (size_t)token * NEXP;
#pragma unroll
        for (int e = 0; e < NEXP; ++e)
            crow[e] = (e == i1) ? p[i1] : ((e == i2) ? p[i2] : 0.f);

        const int s1 = atomicAdd(&counts[i1], 1);
        token_idx[i1 * NBATCH + s1] = token;
        const int s2 = atomicAdd(&counts[i2], 1);
        token_idx[i2 * NBATCH + s2] = token;
    }
}

// ---------------------------------------------------------------------------
// Kernel 2: out[b,o] = sum_e combine[b,e] * b_e[e,o]  (bias base for atomics)
// ---------------------------------------------------------------------------
__global__ __launch_bounds__(256) void bias_base_kernel(
    const float* __restrict__ be, const float* __restrict__ combine,
    float* __restrict__ out) {
    __shared__ float c[NEXP];
    const int token = blockIdx.x;
    if (threadIdx.x < NEXP)
        c[threadIdx.x] = combine[(size_t)token * NEXP + threadIdx.x];
    __syncthreads();
    for (int o = threadIdx.x; o < OUT_DIM; o += 256) {
        float s = 0.f;
#pragma unroll
        for (int e = 0; e < NEXP; ++e) s += c[e] * be[e * OUT_DIM + o];
        out[(size_t)token * OUT_DIM + o] = s;
    }
}

// ---------------------------------------------------------------------------
// Kernel 3: grouped GEMM per expert with gathered rows, f32 WMMA 16x16x4.
// 256 threads = 8 waves (4M x 2N); tile 64(M) x 128(N), K-chunk 16.
// Double-buffered LDS; A staged via GLOBAL_LOAD_ASYNC_TO_LDS_B128 (gathered,
// no VGPR round-trip, ASYNCcnt-tracked); B staged through registers with a
// K-pair-interleaved LDS layout so each WMMA B fragment is one ds_load_b64.
// Symmetric 2-chunk loop body so accumulators never get copied.
// ---------------------------------------------------------------------------
#define A_LD   20    // floats/row: float4-aligned, conflict-free fragment reads
#define BP_ROW 288   // floats per k-pair row = 2*(TN+16); 1152B stride = 32-bank shift

__global__ __launch_bounds__(256) void moe_gemm_kernel(
    const float* __restrict__ x, const float* __restrict__ We,
    const float* __restrict__ combine, const int* __restrict__ counts,
    const int* __restrict__ token_idx, float* __restrict__ out) {
    __shared__ float As[2][TM * A_LD];            // 2 x 5120 B
    __shared__ float Bp[2][(KB / 2) * BP_ROW];    // 2 x 9216 B
    __shared__ int   s_tok[TM];
    __shared__ float s_scale[TM];

    const int e      = blockIdx.z;
    const int cnt    = counts[e];
    const int m_base = blockIdx.y * TM;
    if (m_base >= cnt) return;                    // block-uniform: EXEC stays full
    const int n_base = blockIdx.x * TN;
    const int tid    = threadIdx.x;

    if (tid < TM) {
        const int m = m_base + tid;
        int tok = 0; float sc = 0.f;
        if (m < cnt) {
            tok = token_idx[e * NBATCH + m];
            sc  = combine[(size_t)tok * NEXP + e];
        }
        s_tok[tid]   = tok;
        s_scale[tid] = sc;                        // 0 => padded row adds 0.0
    }
    __syncthreads();

    const int wave = tid >> 5;
    const int lane = tid & 31;
    const int wm   = wave & 3;                    // M wave: 0..3 (16 rows each)
    const int wn   = wave >> 2;                   // N wave: 0..1 (64 cols each)
    const int half = lane >> 4;                   // 0/1
    const int rc   = lane & 15;                   // A row / B,C col within frag
    const int koff = half * 2;

    // A staging: 64 rows x 16 floats; thread t -> row t/4, float4 col (t%4)*4
    const int ar = tid >> 2;
    const int ac = (tid & 3) << 2;
    const float* a_gptr = x + (size_t)s_tok[ar] * IN_DIM + ac;
    // Flat shared pointer low 32 bits == LDS byte offset (flat->LDS truncation)
    const unsigned a_lds0 = (unsigned)(uintptr_t)&As[0][ar * A_LD + ac];
    const unsigned a_lds1 = (unsigned)(uintptr_t)&As[1][ar * A_LD + ac];

    // B staging: two 2x2 micro-tiles per thread: kpairs {t/64, t/64+4}, colpair t%64
    const int bkp = tid >> 6;                     // 0..3
    const int bcp = tid & 63;                     // 0..63
    const float* Wb = We + (size_t)e * IN_DIM * OUT_DIM + n_base + 2 * bcp;

    float2 b_reg[2][2];                           // [j][row-within-pair]

    auto a_async = [&](int k0, int buf) {
        asm volatile("global_load_async_to_lds_b128 %0, %1, off"
                     :: "v"(buf ? a_lds1 : a_lds0), "v"(a_gptr + k0)
                     : "memory");
    };
    auto wait_async = [&]() {
        asm volatile("s_wait_asynccnt 0x0" ::: "memory");
    };
    auto b_prefetch = [&](int k0) {
#pragma unroll
        for (int j = 0; j < 2; ++j) {
            const int r = k0 + 2 * (bkp + 4 * j);
            b_reg[j][0] = *(const float2*)(Wb + (size_t)r * OUT_DIM);
            b_reg[j][1] = *(const float2*)(Wb + (size_t)(r + 1) * OUT_DIM);
        }
    };
    auto b_store = [&](int buf) {
#pragma unroll
        for (int j = 0; j < 2; ++j) {
            const float4 st = {b_reg[j][0].x, b_reg[j][1].x, b_reg[j][0].y, b_reg[j][1].y};
            *(float4*)&Bp[buf][(bkp + 4 * j) * BP_ROW + 4 * bcp] = st;
        }
    };

    v8f c0 = {}, c1 = {}, c2 = {}, c3 = {};

    auto compute = [&](int buf) {
        const float* Ab = &As[buf][(wm * 16 + rc) * A_LD];
        const float* Bb = &Bp[buf][0];
#pragma unroll
        for (int kk = 0; kk < KB; kk += 4) {
            const v2f a = *(const v2f*)(Ab + kk + koff);
            const v2f* bp = (const v2f*)(Bb + (kk / 2 + half) * BP_ROW) + (wn * 64 + rc);
            const v2f b0 = bp[0];
            const v2f b1 = bp[16];
            const v2f b2 = bp[32];
            const v2f b3 = bp[48];
            c0 = __builtin_amdgcn_wmma_f32_16x16x4_f32(false, a, false, b0, (short)0, c0, false, false);
            c1 = __builtin_amdgcn_wmma_f32_16x16x4_f32(false, a, false, b1, (short)0, c1, false, false);
            c2 = __builtin_amdgcn_wmma_f32_16x16x4_f32(false, a, false, b2, (short)0, c2, false, false);
            c3 = __builtin_amdgcn_wmma_f32_16x16x4_f32(false, a, false, b3, (short)0, c3, false, false);
        }
    };

    // ---- prologue: stage chunk 0 into buffer 0 ----
    a_async(0, 0);
    b_prefetch(0);
    b_store(0);
    wait_async();
    __syncthreads();

    // ---- symmetric double-chunk main loop: chunks c (buf0-parity) & c+1 ----
#pragma unroll 1
    for (int c = 0; c < NCHUNK - 2; c += 2) {
        a_async((c + 1) * KB, 1);
        b_prefetch((c + 1) * KB);
        compute(0);
        b_store(1);
        wait_async();
        __syncthreads();

        a_async((c + 2) * KB, 0);
        b_prefetch((c + 2) * KB);
        compute(1);
        b_store(0);
        wait_async();
        __syncthreads();
    }

    // ---- tail: chunk NCHUNK-2 (in buf0) then NCHUNK-1 ----
    a_async((NCHUNK - 1) * KB, 1);
    b_prefetch((NCHUNK - 1) * KB);
    compute(0);
    b_store(1);
    wait_async();
    __syncthreads();
    compute(1);

    // Epilogue: per-row gating scale, scatter-add into out.
#pragma unroll
    for (int v = 0; v < 8; ++v) {
        const int mr = wm * 16 + half * 8 + v;    // local row per C-layout
        const float sc = s_scale[mr];
        float* op = out + (size_t)s_tok[mr] * OUT_DIM + n_base + wn * 64 + rc;
        atomicAdd(op,      c0[v] * sc);
        atomicAdd(op + 16, c1[v] * sc);
        atomicAdd(op + 32, c2[v] * sc);
        atomicAdd(op + 48, c3[v] * sc);
    }
}

// ---------------------------------------------------------------------------
extern "C" void kernel_launch(void* const* d_in, const int* in_sizes, int n_in,
                              void* d_out, int out_size, void* d_ws, size_t ws_size,
                              hipStream_t stream) {
    const float* x  = (const float*)d_in[0];   // [B, I]
    const float* We = (const float*)d_in[1];   // [E, I, O]
    const float* be = (const float*)d_in[2];   // [E, O]
    const float* Wg = (const float*)d_in[3];   // [I, E]
    const float* bg = (const float*)d_in[4];   // [E]
    float* out = (float*)d_out;                // [B, O]

    char* ws = (char*)d_ws;
    float* combine  = (float*)ws;                                   // B*E floats
    int*   counts   = (int*)(ws + (size_t)NBATCH * NEXP * 4);       // E ints
    int*   tok_list = (int*)(ws + (size_t)NBATCH * NEXP * 4 + 64);  // E*B ints

    init_counts_kernel<<<1, 32, 0, stream>>>(counts);
    gating_kernel<<<NBATCH / 8, 256, 0, stream>>>(x, Wg, bg, combine, counts, tok_list);
    bias_base_kernel<<<NBATCH, 256, 0, stream>>>(be, combine, out);
    dim3 grid(OUT_DIM / TN, NBATCH / TM, NEXP);
    moe_gemm_kernel<<<grid, 256, 0, stream>>>(x, We, combine, counts, tok_list, out);
}